// PackedAvgPool1d_19215683682687
// MI455X (gfx1250) — compile-verified
//
#include <hip/hip_runtime.h>
#include <hip/hip_bf16.h>

// PackedAvgPool1d (kernel == stride == 2) for MI455X / gfx1250.
//
// Memory-bound streaming op: ~270MB read + ~135MB write, ~0.17 flop/byte.
// Optimal strategy = saturate 23.3 TB/s HBM with wave32 float4 (b128)
// non-temporal loads/stores; metadata (segment cumsums) staged in LDS.
//
// d_out layout (reference returns a tuple, concatenated flat, fp32):
//   [0 .. new_total*H)                 : y
//   [new_total*H .. +B)                : new_seq_lens (as float)
//   [new_total*H+B .. +B+1)            : new_cu_seq_lens (as float)
//
// d_ws layout (int32):
//   wsi[0]            : H
//   wsi[1 .. B+1]     : cu_seq_lens   (int32 copy, B+1 entries)
//   wsi[B+2 .. 2B+2]  : new_cu        (int32, B+1 entries)

#define TPB 256
#define MAX_B_SMEM 1024   // cap for staging cumsum tables in LDS

// Native clang vector type: required by __builtin_nontemporal_load/store
// (HIP's float4 is a struct and is rejected). Lowers to b128 vmem ops.
typedef float v4f __attribute__((ext_vector_type(4)));

__global__ void pool2_setup_kernel(const int* __restrict__ seq_lens,
                                   const int* __restrict__ cu,
                                   int B, long long x_elems,
                                   float* __restrict__ out_tail,
                                   int* __restrict__ wsi) {
    // Tiny serial setup: B = 64, negligible vs. main kernel.
    if (threadIdx.x == 0 && blockIdx.x == 0) {
        int total = cu[B];
        int H = (total > 0) ? (int)(x_elems / (long long)total) : 0;
        wsi[0] = H;
        int* cu32 = wsi + 1;
        int* ncu  = wsi + 2 + B;
        ncu[0] = 0;
        int acc = 0;
        for (int b = 0; b < B; ++b) {
            cu32[b] = cu[b];
            int len = seq_lens[b];
            int nl  = (len + 1) >> 1;          // ceil(len/2) == new_seq_len
            acc += nl;
            ncu[b + 1] = acc;
            out_tail[b] = (float)nl;           // new_seq_lens (fp32 tail)
        }
        cu32[B] = total;
        for (int b = 0; b <= B; ++b)
            out_tail[B + b] = (float)ncu[b];   // new_cu (fp32 tail)
    }
}

__global__ __launch_bounds__(TPB)
void pool2_main_kernel(const float* __restrict__ x,
                       const int* __restrict__ wsi,
                       float* __restrict__ y,
                       int B, unsigned int nVec) {
    __shared__ int s_meta[2 * (MAX_B_SMEM + 1)];

    const int H = wsi[0];
    const unsigned int vecPerRow = (unsigned int)(H >> 2);

    // Stage both cumsum tables (cu, new_cu) into LDS: 2*(B+1) ints.
    const int* cu;
    const int* ncu;
    if (B <= MAX_B_SMEM) {
        const int nMeta = 2 * (B + 1);
        for (int i = (int)threadIdx.x; i < nMeta; i += (int)blockDim.x)
            s_meta[i] = wsi[1 + i];
        __syncthreads();
        cu  = s_meta;
        ncu = s_meta + (B + 1);
    } else {
        cu  = wsi + 1;
        ncu = wsi + 2 + B;
    }

    const unsigned int v = blockIdx.x * blockDim.x + threadIdx.x;  // float4 index
    if (v >= nVec) return;

    const unsigned int row = v / vecPerRow;          // output row (wave-uniform)
    const unsigned int hv  = v - row * vecPerRow;    // float4 column within row

    // Binary search: seg with new_cu[seg] <= row < new_cu[seg+1]  (LDS reads)
    int lo = 0, hi = B;
    while (hi - lo > 1) {
        int mid = (lo + hi) >> 1;
        if ((unsigned int)ncu[mid] <= row) lo = mid; else hi = mid;
    }
    const int seg = lo;

    const int w    = (int)row - ncu[seg];            // window index in segment
    const int base = cu[seg] + (w << 1);             // first input row of window
    const int rem  = (cu[seg + 1] - cu[seg]) - (w << 1);  // 2, or 1 for odd tail

    const size_t off = (size_t)base * (size_t)H + ((size_t)hv << 2);

    // Streaming, read-once data: non-temporal b128 loads (skip L2 retention).
    v4f a = __builtin_nontemporal_load((const v4f*)(x + off));
    if (rem >= 2) {
        v4f b = __builtin_nontemporal_load((const v4f*)(x + off + (size_t)H));
        a.x = (a.x + b.x) * 0.5f;
        a.y = (a.y + b.y) * 0.5f;
        a.z = (a.z + b.z) * 0.5f;
        a.w = (a.w + b.w) * 0.5f;
    }
    // rem == 1: norm = 1, output = input row unchanged.

    __builtin_nontemporal_store(a, (v4f*)y + v);
}

extern "C" void kernel_launch(void* const* d_in, const int* in_sizes, int n_in,
                              void* d_out, int out_size, void* d_ws, size_t ws_size,
                              hipStream_t stream) {
    const float* x        = (const float*)d_in[0];
    const int*   seq_lens = (const int*)  d_in[1];
    const int*   cu       = (const int*)  d_in[2];
    // d_in[3] = max_seq_len (unused: general path reproduces the identity case)

    const int B = in_sizes[1];                 // 64
    const long long x_elems = (long long)in_sizes[0];

    float* y        = (float*)d_out;
    const int tail  = out_size - (2 * B + 1);  // new_total * H (host-known)
    float* out_tail = y + tail;

    int* wsi = (int*)d_ws;                     // needs (2B+3)*4 bytes (~0.5 KB)

    pool2_setup_kernel<<<1, 64, 0, stream>>>(seq_lens, cu, B, x_elems, out_tail, wsi);

    const unsigned int nVec = (unsigned int)(tail >> 2);   // H is a multiple of 4
    const unsigned int blocks = (nVec + TPB - 1) / TPB;
    if (blocks > 0) {
        pool2_main_kernel<<<blocks, TPB, 0, stream>>>(x, wsi, y, B, nVec);
    }
}